// GCNLayer_28123445854706
// MI455X (gfx1250) — compile-verified
//
#include <hip/hip_runtime.h>
#include <math.h>

#define EPSF 1e-7f
#define PROJ_EPSF 1e-5f
#define MAX_TANH_ARGF 15.0f
#define CRW 0.1f
#define DIM 128

typedef __attribute__((ext_vector_type(2))) float v2f;
typedef __attribute__((ext_vector_type(8))) float v8f;
typedef __attribute__((ext_vector_type(4))) unsigned int v4u;
typedef __attribute__((ext_vector_type(8))) unsigned int v8u;

__device__ __forceinline__ float waveSum(float v) {
    v += __shfl_xor(v, 16, 32);
    v += __shfl_xor(v, 8, 32);
    v += __shfl_xor(v, 4, 32);
    v += __shfl_xor(v, 2, 32);
    v += __shfl_xor(v, 1, 32);
    return v;
}

__device__ __forceinline__ float dot4(float4 a, float4 b) {
    return a.x * b.x + a.y * b.y + a.z * b.z + a.w * b.w;
}
__device__ __forceinline__ float4 scl4(float s, float4 a) {
    float4 r; r.x = s * a.x; r.y = s * a.y; r.z = s * a.z; r.w = s * a.w; return r;
}
__device__ __forceinline__ float4 axpby4(float s, float4 a, float t, float4 b) {
    float4 r;
    r.x = s * a.x + t * b.x; r.y = s * a.y + t * b.y;
    r.z = s * a.z + t * b.z; r.w = s * a.w + t * b.w;
    return r;
}

__device__ __forceinline__ void atomicMaxF(float* addr, float val) {
    unsigned int* a = (unsigned int*)addr;
    unsigned int old = *a;
    while (__uint_as_float(old) < val) {
        unsigned int assumed = old;
        old = atomicCAS(a, assumed, __float_as_uint(val));
        if (old == assumed) break;
    }
}

// ---------------- Stage 0: transpose W so B-operand pairs are contiguous ----------------
__global__ void k_transW(const float* __restrict__ W, float* __restrict__ Wt) {
    int idx = blockIdx.x * blockDim.x + threadIdx.x;  // DIM*DIM threads
    int r = idx >> 7, c = idx & (DIM - 1);
    Wt[(size_t)c * DIM + r] = W[idx];  // Wt[col][k] = W[k][col]
}

// ---------------- Stage 1: log_map_zero (wave per row) ----------------
__global__ void k_logmap(const float* __restrict__ x, float* __restrict__ out, int N) {
    int wid = (blockIdx.x * blockDim.x + threadIdx.x) >> 5;
    int lane = threadIdx.x & 31;
    if (wid >= N) return;
    float4 v = ((const float4*)(x + (size_t)wid * DIM))[lane];
    float n = sqrtf(waveSum(dot4(v, v)));
    float nc = fminf(fmaxf(n, EPSF), 1.0f - PROJ_EPSF);
    float scale = atanhf(nc) / fmaxf(n, EPSF);
    ((float4*)(out + (size_t)wid * DIM))[lane] = scl4(scale, v);
}

// ---------------- Stage 2: h = ents_tan @ W via TDM + WMMA f32 16x16x4 ----------------
__global__ void __launch_bounds__(256) k_gemm_wmma(const float* __restrict__ A,
                                                   const float* __restrict__ Wt,
                                                   float* __restrict__ H, int N) {
    __shared__ float sA[16 * DIM];  // 8 KB A tile, filled by the Tensor Data Mover
    const int t = threadIdx.x;
    const int r0 = blockIdx.x * 16;

    if (t < 32) {
        // Build Tensor DMA descriptor (D#): 1-D tile of 16*DIM f32, contiguous.
        // OOB rows (r0+16 > N) read as zero via tensor_dim0 clamp.
        unsigned long long gaddr =
            (unsigned long long)(const void*)(A + (size_t)r0 * DIM);
        unsigned int ldsoff = (unsigned int)(unsigned long long)(void*)sA;  // flat LDS addr[31:0]
        long long avail = ((long long)N - (long long)r0) * DIM;
        unsigned int td0 = (avail >= 16 * DIM) ? (16u * DIM) : (unsigned int)avail;

        v4u g0;
        g0.x = 1u;                                       // count=1, user descriptor
        g0.y = ldsoff;                                   // lds_addr
        g0.z = (unsigned int)gaddr;                      // global_addr[31:0]
        g0.w = (unsigned int)(gaddr >> 32) | (2u << 30); // global_addr[56:32] | type=2

        v8u g1;
        g1.s0 = 0x00020000u;                 // workgroup_mask=0, data_size=2 (4B)
        g1.s1 = (td0 & 0xffffu) << 16;       // tensor_dim0[15:0]
        g1.s2 = (td0 >> 16) & 0xffffu;       // tensor_dim0[31:16], tensor_dim1=0
        g1.s3 = (16u * DIM) << 16;           // tile_dim0 = 2048
        g1.s4 = 0u;                          // tile_dim1=0, tile_dim2=0
        g1.s5 = 16u * DIM;                   // tensor_dim0_stride[31:0]
        g1.s6 = 0u;
        g1.s7 = 0u;

        v4u gz = {0u, 0u, 0u, 0u};
        asm volatile("tensor_load_to_lds %0, %1, %2, %3"
                     :
                     : "s"(g0), "s"(g1), "s"(gz), "s"(gz)
                     : "memory");
        __builtin_amdgcn_s_wait_tensorcnt(0);
    }
    __syncthreads();

    const int wave = t >> 5;          // column tile 0..7
    const int lane = t & 31;
    const int half = lane >> 4;       // 0 or 1
    const int lm = lane & 15;
    const int col = wave * 16 + lm;

    v8f acc = {};
#pragma unroll
    for (int k0 = 0; k0 < DIM; k0 += 4) {
        const int ka = k0 + half * 2;
        // A 16x4 layout: lane L holds A[L%16][2*(L/16)+i] in vgpr i
        v2f a = *(const v2f*)(sA + lm * DIM + ka);
        // B 4x16 layout: lane L holds B[2*(L/16)+i][L%16] in vgpr i
        v2f b = *(const v2f*)(Wt + (size_t)col * DIM + ka);
        acc = __builtin_amdgcn_wmma_f32_16x16x4_f32(false, a, false, b,
                                                    (short)0, acc, false, false);
    }
    // C/D layout: vgpr v holds row M = v + 8*half, N = lane%16
#pragma unroll
    for (int v = 0; v < 8; ++v) {
        int row = r0 + v + 8 * half;
        if (row < N) H[(size_t)row * DIM + col] = acc[v];
    }
}

// ---------------- Stage 3: init accumulators ----------------
__global__ void k_init(float* __restrict__ near_ents, float* __restrict__ near_rels,
                       float* __restrict__ rowmax, float* __restrict__ denom,
                       size_t ND, int N) {
    size_t i = (size_t)blockIdx.x * blockDim.x + threadIdx.x;
    if (i < ND) { near_ents[i] = 0.f; near_rels[i] = 0.f; }
    if (i < (size_t)N) { rowmax[i] = -3.402823466e38f; denom[i] = 0.f; }
}

// ---------------- Stage 4: edge scores + segment max (wave per edge) ----------------
__global__ void k_scores(const float* __restrict__ h, const int* __restrict__ er,
                         const int* __restrict__ ec, float* __restrict__ scores,
                         float* __restrict__ rowmax, int E) {
    int wid = (blockIdx.x * blockDim.x + threadIdx.x) >> 5;
    int lane = threadIdx.x & 31;
    if (wid >= E) return;
    int r = er[wid], c = ec[wid];
    float4 a = ((const float4*)(h + (size_t)r * DIM))[lane];
    float4 b = ((const float4*)(h + (size_t)c * DIM))[lane];
    float s = waveSum(dot4(a, b));
    if (lane == 0) {
        scores[wid] = s;
        atomicMaxF(&rowmax[r], s);
    }
}

// ---------------- Stage 5: exp + segment sum (thread per edge) ----------------
__global__ void k_expdenom(const int* __restrict__ er, float* __restrict__ scores,
                           const float* __restrict__ rowmax, float* __restrict__ denom, int E) {
    int e = blockIdx.x * blockDim.x + threadIdx.x;
    if (e >= E) return;
    int r = er[e];
    float v = expf(scores[e] - rowmax[r]);
    scores[e] = v;  // reuse buffer as e-values
    atomicAdd(&denom[r], v);
}

// ---------------- Stage 6: attention-weighted scatter (wave per edge) ----------------
__global__ void k_scatter_ents(const float* __restrict__ h, const int* __restrict__ er,
                               const int* __restrict__ ec, const float* __restrict__ evals,
                               const float* __restrict__ denom,
                               float* __restrict__ near_ents, int E) {
    int wid = (blockIdx.x * blockDim.x + threadIdx.x) >> 5;
    int lane = threadIdx.x & 31;
    if (wid >= E) return;
    int r = er[wid], c = ec[wid];
    float alpha = evals[wid] / fmaxf(denom[r], EPSF);
    float4 hv = ((const float4*)(h + (size_t)c * DIM))[lane];
    float* dst = near_ents + (size_t)r * DIM + lane * 4;
    atomicAdd(dst + 0, alpha * hv.x);
    atomicAdd(dst + 1, alpha * hv.y);
    atomicAdd(dst + 2, alpha * hv.z);
    atomicAdd(dst + 3, alpha * hv.w);
}

// ---------------- Stage 7: relation scatter-sum (wave per edge) ----------------
__global__ void k_scatter_rels(const float* __restrict__ rels, const int* __restrict__ rr,
                               const int* __restrict__ rv, float* __restrict__ near_rels, int E) {
    int wid = (blockIdx.x * blockDim.x + threadIdx.x) >> 5;
    int lane = threadIdx.x & 31;
    if (wid >= E) return;
    int r = rr[wid], v = rv[wid];
    float4 rvv = ((const float4*)(rels + (size_t)v * DIM))[lane];
    float* dst = near_rels + (size_t)r * DIM + lane * 4;
    atomicAdd(dst + 0, rvv.x);
    atomicAdd(dst + 1, rvv.y);
    atomicAdd(dst + 2, rvv.z);
    atomicAdd(dst + 3, rvv.w);
}

// ---------------- Stage 8: exp-map, projections, Mobius (wave per row) ----------------
__global__ void k_finalize(const float* __restrict__ near_ents, const float* __restrict__ near_rels,
                           const float* __restrict__ nrn, const float* __restrict__ bias,
                           float* __restrict__ out, int N) {
    int wid = (blockIdx.x * blockDim.x + threadIdx.x) >> 5;
    int lane = threadIdx.x & 31;
    if (wid >= N) return;

    float4 ne = ((const float4*)(near_ents + (size_t)wid * DIM))[lane];
    float4 nr = ((const float4*)(near_rels + (size_t)wid * DIM))[lane];
    float inv = CRW / nrn[wid];
    float4 v = axpby4(1.0f, ne, inv, nr);

    // exp_map_zero + projection for v
    float n1 = fmaxf(sqrtf(waveSum(dot4(v, v))), EPSF);
    float sc = tanhf(fminf(n1, MAX_TANH_ARGF)) / n1;
    float4 u = scl4(sc, v);
    float nu = fmaxf(sqrtf(waveSum(dot4(u, u))), EPSF);
    u = scl4(fminf(1.0f, (1.0f - PROJ_EPSF) / nu), u);

    // b = projection(exp_map_zero(bias))
    float4 bv = ((const float4*)bias)[lane];
    float nb = fmaxf(sqrtf(waveSum(dot4(bv, bv))), EPSF);
    float scb = tanhf(fminf(nb, MAX_TANH_ARGF)) / nb;
    float4 b = scl4(scb, bv);
    float nb2 = fmaxf(sqrtf(waveSum(dot4(b, b))), EPSF);
    b = scl4(fminf(1.0f, (1.0f - PROJ_EPSF) / nb2), b);

    // mobius_addition(u, b)
    float uv = waveSum(dot4(u, b));
    float u2 = waveSum(dot4(u, u));
    float v2 = waveSum(dot4(b, b));
    float cu = 1.0f + 2.0f * uv + v2;
    float cb = 1.0f - u2;
    float den = fmaxf(1.0f + 2.0f * uv + u2 * v2, EPSF);
    float4 res = scl4(1.0f / den, axpby4(cu, u, cb, b));

    // final projection
    float nres = fmaxf(sqrtf(waveSum(dot4(res, res))), EPSF);
    res = scl4(fminf(1.0f, (1.0f - PROJ_EPSF) / nres), res);

    ((float4*)(out + (size_t)wid * DIM))[lane] = res;
}

extern "C" void kernel_launch(void* const* d_in, const int* in_sizes, int n_in,
                              void* d_out, int out_size, void* d_ws, size_t ws_size,
                              hipStream_t stream) {
    const float* ents    = (const float*)d_in[0];
    const float* rels    = (const float*)d_in[1];
    const float* Went    = (const float*)d_in[2];
    const float* bias    = (const float*)d_in[3];
    const int*   ent_row = (const int*)d_in[4];
    const int*   ent_col = (const int*)d_in[5];
    const int*   rel_row = (const int*)d_in[6];
    const int*   rel_val = (const int*)d_in[7];
    const float* nrn     = (const float*)d_in[8];

    const int N = in_sizes[0] / DIM;
    const int E = in_sizes[4];
    float* out = (float*)d_out;

    float* ws = (float*)d_ws;
    const size_t ND = (size_t)N * DIM;
    float* tan_ne = ws;            // ents_tan, reused as near_ents after GEMM
    float* h      = ws + ND;
    float* nrels  = ws + 2 * ND;
    float* scores = ws + 3 * ND;   // later holds exp values
    float* rowmax = scores + E;
    float* denom  = rowmax + N;
    float* Wt     = denom + N;     // transposed weight matrix (DIM*DIM)

    const int rowWaveBlocks  = (N + 7) / 8;     // 8 waves of 32 per 256-thread block
    const int edgeWaveBlocks = (E + 7) / 8;

    k_transW<<<(DIM * DIM) / 256, 256, 0, stream>>>(Went, Wt);
    k_logmap<<<rowWaveBlocks, 256, 0, stream>>>(ents, tan_ne, N);
    k_gemm_wmma<<<(N + 15) / 16, 256, 0, stream>>>(tan_ne, Wt, h, N);
    {
        int blocks = (int)((ND + 255) / 256);
        k_init<<<blocks, 256, 0, stream>>>(tan_ne, nrels, rowmax, denom, ND, N);
    }
    k_scores<<<edgeWaveBlocks, 256, 0, stream>>>(h, ent_row, ent_col, scores, rowmax, E);
    k_expdenom<<<(E + 255) / 256, 256, 0, stream>>>(ent_row, scores, rowmax, denom, E);
    k_scatter_ents<<<edgeWaveBlocks, 256, 0, stream>>>(h, ent_row, ent_col, scores, denom, tan_ne, E);
    k_scatter_rels<<<edgeWaveBlocks, 256, 0, stream>>>(rels, rel_row, rel_val, nrels, E);
    k_finalize<<<rowWaveBlocks, 256, 0, stream>>>(tan_ne, nrels, nrn, bias, out, N);
}